// AttentionBlock_59811714564497
// MI455X (gfx1250) — compile-verified
//
#include <hip/hip_runtime.h>
#include <hip/hip_bf16.h>
#include <math.h>

// ---------------------------------------------------------------------------
// AttentionBlock for MI455X (gfx1250, wave32) using fp32 WMMA (16x16x4).
// Pipeline: GroupNorm -> QKV GEMM -> flash attention -> proj GEMM + residual.
// ---------------------------------------------------------------------------

typedef __attribute__((ext_vector_type(2))) float v2f;
typedef __attribute__((ext_vector_type(8))) float v8f;

#define WMMA_F32(a, b, c) \
  __builtin_amdgcn_wmma_f32_16x16x4_f32(false, (a), false, (b), (short)0, (c), false, false)

static constexpr int kB  = 8;
static constexpr int kC  = 512;
static constexpr int kN  = 1024;   // H*W = 32*32
static constexpr int kNH = 8;
static constexpr int kHC = 64;     // C / NH
static constexpr int kG  = 32;     // groups
static constexpr int kCPG = 16;    // channels per group

// ---------------------------------------------------------------------------
// 1) GroupNorm: one block per (batch, group). 16 channels x 1024 pixels.
// ---------------------------------------------------------------------------
__global__ __launch_bounds__(256) void gn_kernel(const float* __restrict__ x,
                                                 const float* __restrict__ w,
                                                 const float* __restrict__ bgn,
                                                 float* __restrict__ h) {
  const int blk = blockIdx.x;          // b*32 + g
  const int b = blk >> 5;
  const int g = blk & 31;
  const int total = kCPG * kN;         // 16384
  const float* xp = x + ((size_t)b * kC + (size_t)g * kCPG) * kN;
  float* hp = h + ((size_t)b * kC + (size_t)g * kCPG) * kN;

  float s = 0.f, s2 = 0.f;
  for (int i = threadIdx.x; i < total; i += 256) {
    float v = xp[i];
    s += v;
    s2 += v * v;
  }
  __shared__ float red0[256];
  __shared__ float red1[256];
  red0[threadIdx.x] = s;
  red1[threadIdx.x] = s2;
  __syncthreads();
  for (int off = 128; off > 0; off >>= 1) {
    if ((int)threadIdx.x < off) {
      red0[threadIdx.x] += red0[threadIdx.x + off];
      red1[threadIdx.x] += red1[threadIdx.x + off];
    }
    __syncthreads();
  }
  const float mean = red0[0] * (1.0f / total);
  const float var  = red1[0] * (1.0f / total) - mean * mean;
  const float rstd = rsqrtf(var + 1e-5f);
  for (int i = threadIdx.x; i < total; i += 256) {
    const int c = g * kCPG + (i >> 10);   // i / kN
    float v = (xp[i] - mean) * rstd;
    hp[i] = v * w[c] + bgn[c];
  }
}

// ---------------------------------------------------------------------------
// 2) fp32 WMMA GEMM: O[b] = A[M,K] x Bm[b][K,Nn] + bias[M] (+ resid[b][M,Nn])
//    One wave computes a 64x64 output tile: 4x4 accumulators of 16x16.
//    A fragment (16x4): lane = M (0..15), half-wave selects K pair.
//    B fragment (4x16): lane = N (0..15), half-wave selects K pair.
//    launch_bounds(128,1): keep all 16 accumulators resident (no spills).
//    HAS_RESID is a compile-time flag -> branch-free epilogue.
// ---------------------------------------------------------------------------
template <bool HAS_RESID>
__global__ __launch_bounds__(128, 1) void
gemm_wmma_kernel(const float* __restrict__ A,
                 const float* __restrict__ Bbase,
                 const float* __restrict__ bias,
                 const float* __restrict__ Rbase,
                 float* __restrict__ Obase,
                 int M, int K, int Nn,
                 long strideB, long strideO, long strideR,
                 int mtiles, int ntiles, int totalWaves) {
  const int lane = threadIdx.x & 31;
  const int wave = threadIdx.x >> 5;
  const int gwave = blockIdx.x * (blockDim.x >> 5) + wave;
  if (gwave >= totalWaves) return;

  const int tilesPerBatch = mtiles * ntiles;
  const int batch = gwave / tilesPerBatch;
  const int t = gwave - batch * tilesPerBatch;
  const int m0 = (t / ntiles) * 64;
  const int n0 = (t % ntiles) * 64;

  const float* __restrict__ Bm = Bbase + (size_t)batch * strideB;
  float* __restrict__ O = Obase + (size_t)batch * strideO;

  const int half = lane >> 4;     // K-pair select
  const int l16  = lane & 15;     // M (A) / N (B,C) index within tile

  v8f acc[4][4];
#pragma unroll
  for (int i = 0; i < 4; ++i)
#pragma unroll
    for (int j = 0; j < 4; ++j) acc[i][j] = (v8f){};

  for (int k0 = 0; k0 < K; k0 += 4) {
    const int ka = k0 + half * 2;
    v2f af[4], bf[4];
#pragma unroll
    for (int ti = 0; ti < 4; ++ti) {
      const size_t row = (size_t)(m0 + ti * 16 + l16);
      af[ti] = *(const v2f*)(A + row * K + ka);        // K, K+1 adjacent
    }
#pragma unroll
    for (int tj = 0; tj < 4; ++tj) {
      const size_t col = (size_t)(n0 + tj * 16 + l16);
      bf[tj].x = Bm[(size_t)ka * Nn + col];
      bf[tj].y = Bm[(size_t)(ka + 1) * Nn + col];
    }
#pragma unroll
    for (int ti = 0; ti < 4; ++ti)
#pragma unroll
      for (int tj = 0; tj < 4; ++tj)
        acc[ti][tj] = WMMA_F32(af[ti], bf[tj], acc[ti][tj]);
  }

  // Epilogue: C/D layout -> VGPR r holds row (r + 8*half), lane l16 = column.
  const float* __restrict__ Rb =
      HAS_RESID ? (Rbase + (size_t)batch * strideR) : nullptr;
#pragma unroll
  for (int ti = 0; ti < 4; ++ti) {
#pragma unroll
    for (int r = 0; r < 8; ++r) {
      const int row = m0 + ti * 16 + r + half * 8;
      const float bv = bias[row];
      const size_t base = (size_t)row * Nn + n0 + l16;
      float* __restrict__ orow = O + base;
#pragma unroll
      for (int tj = 0; tj < 4; ++tj) {
        float vv = acc[ti][tj][r] + bv;
        if (HAS_RESID) vv += Rb[base + tj * 16];
        orow[tj * 16] = vv;
      }
    }
  }
}

// ---------------------------------------------------------------------------
// 3) Flash attention. Block = 8 waves; each wave owns 16 queries of one
//    (batch, head). S^T tiles (M = keys, N = queries) keep softmax in-lane.
//    P^T is relaid through per-wave LDS to feed the O += V * P^T WMMA.
// ---------------------------------------------------------------------------
__global__ __launch_bounds__(256, 1) void
attn_kernel(const float* __restrict__ qkv,
            float* __restrict__ aout) {
  const float scale = 0.125f;   // hc^-0.5 = 1/8
  const int lane = threadIdx.x & 31;
  const int wave = threadIdx.x >> 5;     // 0..7
  const int blk = blockIdx.x;            // B*NH*(N/128) = 512
  const int ib128 = blk & 7;
  const int head = (blk >> 3) & 7;
  const int b = blk >> 6;
  const int i0 = ib128 * 128 + wave * 16;

  const float* __restrict__ q = qkv + (((size_t)b * 3 + 0) * kC + (size_t)head * kHC) * kN;
  const float* __restrict__ k = qkv + (((size_t)b * 3 + 1) * kC + (size_t)head * kHC) * kN;
  const float* __restrict__ v = qkv + (((size_t)b * 3 + 2) * kC + (size_t)head * kHC) * kN;
  float* __restrict__ o = aout + ((size_t)b * kC + (size_t)head * kHC) * kN;

  __shared__ float pshm[8][16][16];      // per-wave P^T tile [j][i]

  const int half = lane >> 4;
  const int l16  = lane & 15;

  // O accumulators: O[c, i] over 4 c-tiles of 16; lane l16 = query column.
  v8f oacc[4];
#pragma unroll
  for (int t = 0; t < 4; ++t) oacc[t] = (v8f){};
  float mrow = -INFINITY;   // running max for query i0+l16
  float lrow = 0.f;         // running denom

  // Preload all q B-fragments (K = channel, 16 steps of 4) — reused 64x.
  v2f qf[16];
#pragma unroll
  for (int s = 0; s < 16; ++s) {
    const int c = s * 4 + half * 2;
    qf[s].x = q[(size_t)c * kN + i0 + l16];
    qf[s].y = q[(size_t)(c + 1) * kN + i0 + l16];
  }

  for (int jt = 0; jt < 64; ++jt) {
    const int jb = jt * 16;

    // S^T tile = K_tile^T x Q_tile  (M = key j, N = query i, K = channel)
    v8f st = (v8f){};
#pragma unroll
    for (int s = 0; s < 16; ++s) {
      const int c = s * 4 + half * 2;
      v2f af;
      af.x = k[(size_t)c * kN + jb + l16];
      af.y = k[(size_t)(c + 1) * kN + jb + l16];
      st = WMMA_F32(af, qf[s], st);
    }

    // Online softmax: lane holds 8 key-scores of one query; pair-lane (xor 16)
    // holds the other 8 keys of the same query.
    float sv[8];
    float tmax = -INFINITY;
#pragma unroll
    for (int r = 0; r < 8; ++r) {
      sv[r] = st[r] * scale;
      tmax = fmaxf(tmax, sv[r]);
    }
    tmax = fmaxf(tmax, __shfl_xor(tmax, 16));
    const float mnew = fmaxf(mrow, tmax);
    const float alpha = __expf(mrow - mnew);
    float psum = 0.f;
#pragma unroll
    for (int r = 0; r < 8; ++r) {
      sv[r] = __expf(sv[r] - mnew);
      psum += sv[r];
    }
    psum += __shfl_xor(psum, 16);
    lrow = lrow * alpha + psum;
    mrow = mnew;
#pragma unroll
    for (int t = 0; t < 4; ++t)
#pragma unroll
      for (int r = 0; r < 8; ++r) oacc[t][r] *= alpha;

    // Relayout P^T through LDS (C-layout -> B-operand layout).
    __syncthreads();
#pragma unroll
    for (int r = 0; r < 8; ++r) pshm[wave][r + half * 8][l16] = sv[r];
    __syncthreads();

    // O[c, i] += V[c, jb..jb+15] * P^T[j, i]   (K = key j, 4 steps of 4)
#pragma unroll
    for (int s = 0; s < 4; ++s) {
      const int j = s * 4 + half * 2;
      v2f bf;
      bf.x = pshm[wave][j][l16];
      bf.y = pshm[wave][j + 1][l16];
#pragma unroll
      for (int t = 0; t < 4; ++t) {
        const int c = t * 16 + l16;
        v2f af;
        af.x = v[(size_t)c * kN + jb + j];
        af.y = v[(size_t)c * kN + jb + j + 1];
        oacc[t] = WMMA_F32(af, bf, oacc[t]);
      }
    }
  }

  // Finalize: divide by softmax denominator, write O[c, i].
  const float inv = 1.f / lrow;
#pragma unroll
  for (int t = 0; t < 4; ++t) {
#pragma unroll
    for (int r = 0; r < 8; ++r) {
      const int c = t * 16 + r + half * 8;
      o[(size_t)c * kN + i0 + l16] = oacc[t][r] * inv;
    }
  }
}

// ---------------------------------------------------------------------------
// Launch
// ---------------------------------------------------------------------------
extern "C" void kernel_launch(void* const* d_in, const int* in_sizes, int n_in,
                              void* d_out, int out_size, void* d_ws, size_t ws_size,
                              hipStream_t stream) {
  const float* x    = (const float*)d_in[0];
  const float* gnw  = (const float*)d_in[1];
  const float* gnb  = (const float*)d_in[2];
  const float* qkvw = (const float*)d_in[3];   // [1536, 512]
  const float* qkvb = (const float*)d_in[4];   // [1536]
  const float* pw   = (const float*)d_in[5];   // [512, 512]
  const float* pb   = (const float*)d_in[6];   // [512]
  float* out = (float*)d_out;

  const size_t perBatch = (size_t)kC * kN;         // 524288
  float* h    = (float*)d_ws;                      // [B, C, N]
  float* qkv  = h + (size_t)kB * perBatch;         // [B, 3C, N]
  float* aout = qkv + (size_t)kB * 3 * perBatch;   // [B, C, N]

  // 1) GroupNorm
  gn_kernel<<<kB * kG, 256, 0, stream>>>(x, gnw, gnb, h);

  // 2) QKV GEMM: M=1536, K=512, N=1024 per batch. 24x16 wave-tiles x 8 batches.
  {
    const int mtiles = (3 * kC) / 64, ntiles = kN / 64;
    const int totalWaves = kB * mtiles * ntiles;   // 3072
    const int blocks = (totalWaves + 3) / 4;
    gemm_wmma_kernel<false><<<blocks, 128, 0, stream>>>(
        qkvw, h, qkvb, nullptr, qkv,
        3 * kC, kC, kN,
        (long)perBatch, (long)(3 * perBatch), 0L,
        mtiles, ntiles, totalWaves);
  }

  // 3) Attention: 8 waves/block, 128 queries/block.
  attn_kernel<<<kB * kNH * (kN / 128), 256, 0, stream>>>(qkv, aout);

  // 4) Projection GEMM + bias + residual: M=512, K=512, N=1024.
  {
    const int mtiles = kC / 64, ntiles = kN / 64;
    const int totalWaves = kB * mtiles * ntiles;   // 1024
    const int blocks = (totalWaves + 3) / 4;
    gemm_wmma_kernel<true><<<blocks, 128, 0, stream>>>(
        pw, aout, pb, x, out,
        kC, kC, kN,
        (long)perBatch, (long)perBatch, (long)perBatch,
        mtiles, ntiles, totalWaves);
  }
}